// Projector_44212393345207
// MI455X (gfx1250) — compile-verified
//
#include <hip/hip_runtime.h>

typedef __attribute__((ext_vector_type(2))) float v2f;
typedef __attribute__((ext_vector_type(4))) float v4f;
typedef __attribute__((ext_vector_type(8))) float v8f;

#define CC 32
#define HH 512
#define WW 512
#define FD 16
#define BB 65536

// One wave (32 lanes) handles 16 positions x 32 cameras.
// Projection done with 6x V_WMMA_F32_16X16X4_F32:
//   A  (16x4)  = homogeneous positions [x y z 1]
//   B_r(4x16)  = column n holds row r of M[camBase+n]  (r = 0,1,2)
//   D_r(16x16) = D_r[p][c] = dot(M[c][r], homo[p])
// C/D layout: VGPR v, lanes 0-15 -> p = v, lanes 16-31 -> p = v+8; column = lane&15.
// => u, v, w for one (p,c) pair live in the SAME lane & VGPR slot: epilogue is lane-local.
__global__ __launch_bounds__(256) void Projector_44212393345207_kernel(
    const float* __restrict__ pos,    // [B,3]
    const float* __restrict__ img,    // [C,H,W,3]
    const float* __restrict__ feat,   // [C,H,W,16]
    const float* __restrict__ Mm,     // [C,3,4]
    float* __restrict__ outImg,       // [B,C,3]
    float* __restrict__ outFeat,      // [B,C,16]
    float* __restrict__ outMask)      // [B,C]
{
    const int lane = threadIdx.x & 31;
    const int wave = threadIdx.x >> 5;
    const int base = blockIdx.x * 128 + wave * 16;   // 16 positions per wave
    if (base >= BB) return;                          // uniform per wave: EXEC stays all-ones

    const int half = lane >> 4;      // 0 = lanes 0-15, 1 = lanes 16-31
    const int l16  = lane & 15;

    // ---- A operand: f32 16x4 A-layout (lanes 0-15: K=0,1 ; lanes 16-31: K=2,3) ----
    const int p = base + l16;
    const float px = pos[3 * p + 0];
    const float py = pos[3 * p + 1];
    const float pz = pos[3 * p + 2];
    v2f A;
    A.x = half ? pz   : px;
    A.y = half ? 1.0f : py;

    // ---- B operands: 4x16, column n = M[camBase+n][r][0..3] (mirrored layout) ----
    v2f Bm[2][3];
    #pragma unroll
    for (int g = 0; g < 2; ++g) {
        const int cam = g * 16 + l16;
        #pragma unroll
        for (int r = 0; r < 3; ++r) {
            const float* mrow = Mm + cam * 12 + r * 4;
            Bm[g][r].x = half ? mrow[2] : mrow[0];
            Bm[g][r].y = half ? mrow[3] : mrow[1];
        }
    }

    // ---- 6x WMMA f32 16x16x4 ----
    v8f acc[2][3];
    #pragma unroll
    for (int g = 0; g < 2; ++g) {
        #pragma unroll
        for (int r = 0; r < 3; ++r) {
            v8f z = {};
            acc[g][r] = __builtin_amdgcn_wmma_f32_16x16x4_f32(
                /*neg_a=*/false, A, /*neg_b=*/false, Bm[g][r],
                /*c_mod=*/(short)0, z, /*reuse_a=*/false, /*reuse_b=*/false);
        }
    }

    // ---- lane-local epilogue: divide, truncate, mask, gather, stream out ----
    #pragma unroll
    for (int g = 0; g < 2; ++g) {
        const int c = g * 16 + l16;
        #pragma unroll
        for (int r = 0; r < 8; ++r) {
            const int prow = base + r + 8 * half;
            const float u = acc[g][0][r];
            const float v = acc[g][1][r];
            const float w = acc[g][2][r];
            float fx = u / w;                       // ic0 (row, compared vs H)
            float fy = v / w;                       // ic1 (col, compared vs W)
            fx = fminf(fmaxf(fx, -1073741824.0f), 1073741824.0f);
            fy = fminf(fmaxf(fy, -1073741824.0f), 1073741824.0f);
            const int ix = (int)fx;                 // trunc toward zero == astype(int32)
            const int iy = (int)fy;
            const bool m = (ix >= 0) & (iy >= 0) & (iy < WW) & (ix < HH);
            const int jx = m ? ix : 0;
            const int jy = m ? iy : 0;

            const int pix = (c * HH + jx) * WW + jy;
            const float* ip = img + pix * 3;
            const float i0 = ip[0], i1 = ip[1], i2 = ip[2];
            const v4f* fp = (const v4f*)(feat + pix * FD);
            const v4f f0 = fp[0], f1 = fp[1], f2 = fp[2], f3 = fp[3];

            const int o = prow * CC + c;
            float* oi = outImg + o * 3;
            __builtin_nontemporal_store(i0, oi + 0);
            __builtin_nontemporal_store(i1, oi + 1);
            __builtin_nontemporal_store(i2, oi + 2);
            v4f* of = (v4f*)(outFeat + o * FD);
            __builtin_nontemporal_store(f0, of + 0);
            __builtin_nontemporal_store(f1, of + 1);
            __builtin_nontemporal_store(f2, of + 2);
            __builtin_nontemporal_store(f3, of + 3);
            __builtin_nontemporal_store(m ? 1.0f : 0.0f, outMask + o);
        }
    }
}

extern "C" void kernel_launch(void* const* d_in, const int* in_sizes, int n_in,
                              void* d_out, int out_size, void* d_ws, size_t ws_size,
                              hipStream_t stream) {
    (void)in_sizes; (void)n_in; (void)d_ws; (void)ws_size; (void)out_size;
    const float* pos  = (const float*)d_in[0];   // [B,3]
    const float* img  = (const float*)d_in[1];   // [C,H,W,3]
    const float* feat = (const float*)d_in[2];   // [C,H,W,16]
    const float* Mm   = (const float*)d_in[3];   // [C,3,4]

    float* out     = (float*)d_out;
    float* outImg  = out;                          // B*C*3
    float* outFeat = out + (size_t)BB * CC * 3;    // B*C*16
    float* outMask = out + (size_t)BB * CC * 19;   // B*C

    const int posPerBlock = 128;                   // 8 waves x 16 positions
    const int grid = (BB + posPerBlock - 1) / posPerBlock;
    Projector_44212393345207_kernel<<<grid, 256, 0, stream>>>(
        pos, img, feat, Mm, outImg, outFeat, outMask);
}